// glocapBlockFuzzy_42631845380378
// MI455X (gfx1250) — compile-verified
//
#include <hip/hip_runtime.h>
#include <math.h>

typedef __attribute__((ext_vector_type(2))) float v2f;
typedef __attribute__((ext_vector_type(8))) float v8f;

#define NN    32
#define CIN   32
#define DDIM  16
#define WHM   196          // W*H = 14*14, equals capsule index m
#define NCLS  10
#define NBLK  1568         // CIN * (WHM/4) blocks of 4 capsules
#define EPSF  1e-6f
#define LAMF  0.01f

// MODE 0: accumulate sum_r[o]        (r computed against gvA)
// MODE 1: accumulate gvNew[o][d]     (coeff = r/(sumr+eps), weighted V)
// MODE 2: accumulate sigma[o]        (coeff from gvA/sumr, distance to gvB)
template <int MODE>
__device__ __forceinline__ void sweep_pass(
    const float* __restrict__ lp, const float* __restrict__ wgt,
    int n, int wave, int lane, float scale,
    float (*gvA)[16], float (*gvB)[16],
    float* sumr, float* sigma, float (*gvNew)[16],
    float (*sV)[NCLS][16])
{
    const int g    = lane & 7;       // class-slot within 8-lane group
    const int mg   = lane >> 3;      // capsule (0..3) handled by this group
    const int o0   = g;
    const int o1   = g + 8;
    const bool has1 = (g < 2);       // 10 classes over 8 lanes: lanes 0,1 take o+8

    // hoist per-sweep constants into registers
    float ga0[16], ga1[16];
#pragma unroll
    for (int d = 0; d < 16; ++d) {
        ga0[d] = gvA[o0][d];
        ga1[d] = has1 ? gvA[o1][d] : 0.0f;
    }
    float gb0[16], gb1[16];
    if (MODE == 2) {
#pragma unroll
        for (int d = 0; d < 16; ++d) {
            gb0[d] = gvB[o0][d];
            gb1[d] = has1 ? gvB[o1][d] : 0.0f;
        }
    }
    float S0e = 1.0f, S1e = 1.0f;
    if (MODE >= 1) {
        S0e = sumr[o0] + EPSF;
        S1e = has1 ? (sumr[o1] + EPSF) : 1.0f;
    }

    float accA0 = 0.0f, accA1 = 0.0f;   // sum_r or sigma partials
    float accG0[16], accG1[16];         // gv partials (MODE 1)
    if (MODE == 1) {
#pragma unroll
        for (int d = 0; d < 16; ++d) { accG0[d] = 0.0f; accG1[d] = 0.0f; }
    }

    // WMMA lane decode (V_WMMA_F32_16X16X4_F32):
    // A 16x4 f32: lane L -> row M=L&15; VGPR0=A[M,K0], VGPR1=A[M,K0+1], K0=2*(L>>4)
    // B 4x16 f32: lane L -> col N=L&15; VGPR0=B[K0,N], VGPR1=B[K0+1,N]
    // D 16x16:    VGPR v -> M = v + 8*(L>=16), N = L&15
    const int rowA   = lane & 15;
    const int K0     = (lane >> 4) * 2;
    const int iA     = rowA & 3;        // i index within capsule (d = i*4+k)
    const int mA     = rowA >> 2;       // capsule within block (0..3)
    const int osubB  = (lane & 15) >> 2;
    const int kB     = lane & 3;
    const int halfHi = lane >> 4;

    for (int b = wave; b < NBLK; b += 8) {
        int c  = b / 49;
        int mb = b - c * 49;
        int m0 = mb * 4;

        // A tile: lp[n,c,m0+mA,iA,j] = l[n, c, iA*4+j, wh=m]
        v2f A;
        {
            size_t off = ((size_t)n * CIN + c) * (DDIM * WHM)
                       + (size_t)(iA * 4 + K0) * WHM + (size_t)(m0 + mA);
            A.x = lp[off];
            A.y = lp[off + WHM];
        }

#pragma unroll
        for (int t = 0; t < 3; ++t) {
            int o = 4 * t + osubB;
            v2f B;
            if (o < NCLS) {
                size_t wo = ((size_t)(c * NCLS + o) * 4 + K0) * 4 + kB;
                B.x = wgt[wo];
                B.y = wgt[wo + 4];
            } else {
                B.x = 0.0f; B.y = 0.0f;
            }
            v8f Cz = {0.f, 0.f, 0.f, 0.f, 0.f, 0.f, 0.f, 0.f};
            v8f Dm = __builtin_amdgcn_wmma_f32_16x16x4_f32(
                false, A, false, B, (short)0, Cz, false, false);
            // stage D -> LDS: V[m, o, i*4+k]
            if (o < NCLS) {
#pragma unroll
                for (int v = 0; v < 8; ++v) {
                    int Mrow = v + 8 * halfHi;
                    sV[Mrow >> 2][o][(Mrow & 3) * 4 + kB] = Dm[v];
                }
            }
        }
        // wave-local: wait for own LDS stores before cross-lane reads
        asm volatile("s_wait_dscnt 0x0" ::: "memory");

        // ---- routing math: 8-lane group mg consumes capsule mg ----
        float v0[16], v1[16];
        float rn0 = EPSF, rn1 = EPSF;
#pragma unroll
        for (int d = 0; d < 16; ++d) {
            float vv = sV[mg][o0][d];
            v0[d] = vv;
            float df = vv - ga0[d];
            rn0 += df * df;
        }
        if (has1) {
#pragma unroll
            for (int d = 0; d < 16; ++d) {
                float vv = sV[mg][o1][d];
                v1[d] = vv;
                float df = vv - ga1[d];
                rn1 += df * df;
            }
        }
        // r_d = sum over all 10 classes of 1/r_n  (reduce across 8-lane group)
        float invp = 1.0f / rn0 + (has1 ? 1.0f / rn1 : 0.0f);
        invp += __shfl_xor(invp, 1, 8);
        invp += __shfl_xor(invp, 2, 8);
        invp += __shfl_xor(invp, 4, 8);
        const float rd = invp;

        float q0 = 1.0f / (rn0 * rd);
        float r0 = q0 * q0 * scale;      // M=2 -> square; scale=0.1 on iter 0
        float r1 = 0.0f;
        if (has1) { float q1 = 1.0f / (rn1 * rd); r1 = q1 * q1 * scale; }

        if (MODE == 0) {
            accA0 += r0;
            accA1 += r1;
        } else if (MODE == 1) {
            float c0 = r0 / S0e;
#pragma unroll
            for (int d = 0; d < 16; ++d) accG0[d] += c0 * v0[d];
            if (has1) {
                float c1 = r1 / S1e;
#pragma unroll
                for (int d = 0; d < 16; ++d) accG1[d] += c1 * v1[d];
            }
        } else {
            float c0 = r0 / S0e;
            float d2 = 0.0f;
#pragma unroll
            for (int d = 0; d < 16; ++d) { float df = v0[d] - gb0[d]; d2 += df * df; }
            accA0 += c0 * d2;
            if (has1) {
                float c1 = r1 / S1e;
                float d2b = 0.0f;
#pragma unroll
                for (int d = 0; d < 16; ++d) { float df = v1[d] - gb1[d]; d2b += df * df; }
                accA1 += c1 * d2b;
            }
        }
    }

    // one flush per lane per sweep (ds_add_f32)
    if (MODE == 0) {
        atomicAdd(&sumr[o0], accA0);
        if (has1) atomicAdd(&sumr[o1], accA1);
    } else if (MODE == 1) {
#pragma unroll
        for (int d = 0; d < 16; ++d) atomicAdd(&gvNew[o0][d], accG0[d]);
        if (has1) {
#pragma unroll
            for (int d = 0; d < 16; ++d) atomicAdd(&gvNew[o1][d], accG1[d]);
        }
    } else {
        atomicAdd(&sigma[o0], accA0);
        if (has1) atomicAdd(&sigma[o1], accA1);
    }
}

__global__ __launch_bounds__(256) void glocap_fuzzy_route_kernel(
    const float* __restrict__ l, const float* __restrict__ g,
    const float* __restrict__ wgt, const float* __restrict__ beta_a,
    float* __restrict__ out)
{
    __shared__ float s_gvA[NCLS][16];
    __shared__ float s_gvB[NCLS][16];
    __shared__ float s_sumr[NCLS];
    __shared__ float s_sigma[NCLS];
    __shared__ float s_V[8][4][NCLS][16];   // per-wave staging of a 4-capsule block

    const int n    = blockIdx.x;
    const int tid  = threadIdx.x;
    const int wave = tid >> 5;
    const int lane = tid & 31;

    if (tid < NCLS * 16) {
        int o = tid >> 4, d = tid & 15;
        s_gvA[o][d] = g[(size_t)n * NCLS * 16 + tid];   // gv starts at g0
        s_gvB[o][d] = 0.0f;
    }
    if (tid < NCLS) { s_sumr[tid] = 0.0f; s_sigma[tid] = 0.0f; }
    __syncthreads();

    for (int it = 0; it < 3; ++it) {
        float scale = (it == 0) ? (1.0f / (float)NCLS) : 1.0f;  // iter0: r = a_in/10
        sweep_pass<0>(l, wgt, n, wave, lane, scale,
                      s_gvA, s_gvB, s_sumr, s_sigma, s_gvB, s_V[wave]);
        __syncthreads();
        sweep_pass<1>(l, wgt, n, wave, lane, scale,
                      s_gvA, s_gvB, s_sumr, s_sigma, s_gvB, s_V[wave]);
        __syncthreads();
        if (it < 2) {   // advance: gvA <- gvB, clear accumulators
            if (tid < NCLS * 16) {
                int o = tid >> 4, d = tid & 15;
                s_gvA[o][d] = s_gvB[o][d];
                s_gvB[o][d] = 0.0f;
            }
            if (tid < NCLS) s_sumr[tid] = 0.0f;
            __syncthreads();
        }
    }
    // after it=2: gvA holds gv2 (for coeff2), gvB holds gv3, sumr holds S2
    sweep_pass<2>(l, wgt, n, wave, lane, 1.0f,
                  s_gvA, s_gvB, s_sumr, s_sigma, s_gvB, s_V[wave]);
    __syncthreads();

    if (tid < NCLS) {
        float x = LAMF * (beta_a[tid] - 0.5f * logf(s_sigma[tid]));
        out[(size_t)n * NCLS + tid] = 1.0f / (1.0f + expf(-x));
    }
    if (tid < NCLS * 16) {
        out[(size_t)NN * NCLS + (size_t)n * NCLS * 16 + tid] =
            s_gvB[tid >> 4][tid & 15];
    }
}

extern "C" void kernel_launch(void* const* d_in, const int* in_sizes, int n_in,
                              void* d_out, int out_size, void* d_ws, size_t ws_size,
                              hipStream_t stream)
{
    const float* l      = (const float*)d_in[0];
    const float* g      = (const float*)d_in[1];
    const float* wgt    = (const float*)d_in[2];
    const float* beta_a = (const float*)d_in[3];
    float* out = (float*)d_out;
    glocap_fuzzy_route_kernel<<<dim3(NN), dim3(256), 0, stream>>>(l, g, wgt, beta_a, out);
    (void)in_sizes; (void)n_in; (void)out_size; (void)d_ws; (void)ws_size;
}